// MoE_60112362275422
// MI455X (gfx1250) — compile-verified
//
#include <hip/hip_runtime.h>
#include <hip/hip_bf16.h>
#include <math.h>

typedef float v2f __attribute__((ext_vector_type(2)));
typedef float v8f __attribute__((ext_vector_type(8)));

#define NTOK 1024
#define DDIM 1024
#define HDIM 1024
#define NEXP 16
#define XS 1028   // padded LDS row stride for X tile (stride%64==4 -> conflict-free b64 reads)
#define HS 132    // padded LDS row stride for h chunk

// ---------------- init ----------------
__global__ void moe_init_kernel(float* y, float* imp, float* loadv, int* cnt, float* out) {
    size_t i = (size_t)blockIdx.x * blockDim.x + threadIdx.x;
    size_t n = (size_t)NTOK * DDIM;
    size_t stride = (size_t)gridDim.x * blockDim.x;
    for (size_t j = i; j < n; j += stride) y[j] = 0.0f;
    if (blockIdx.x == 0 && threadIdx.x < NEXP) {
        imp[threadIdx.x] = 0.0f;
        loadv[threadIdx.x] = 0.0f;
        cnt[threadIdx.x] = 0;
    }
    if (blockIdx.x == 0 && threadIdx.x == 0) { out[640] = 0.0f; out[641] = 0.0f; }
}

// ---------------- gating: one wave (32 lanes) per token ----------------
__global__ void moe_gate_kernel(const float* __restrict__ x, const float* __restrict__ wg,
                                float* imp, float* loadv, int* cnt, int* etok, float* egate) {
    int wave = threadIdx.x >> 5;
    int lane = threadIdx.x & 31;
    int t = blockIdx.x * 8 + wave;
    if (t >= NTOK) return;

    float acc[NEXP];
#pragma unroll
    for (int e = 0; e < NEXP; e++) acc[e] = 0.0f;
    const float* xr = x + (size_t)t * DDIM;
    for (int d = lane; d < DDIM; d += 32) {
        float xv = xr[d];
        const float* wr = wg + (size_t)d * NEXP;
#pragma unroll
        for (int e = 0; e < NEXP; e++) acc[e] += xv * wr[e];
    }
#pragma unroll
    for (int e = 0; e < NEXP; e++) {
        for (int off = 16; off > 0; off >>= 1) acc[e] += __shfl_down(acc[e], off, 32);
    }
    if (lane == 0) {
        bool used[NEXP];
#pragma unroll
        for (int e = 0; e < NEXP; e++) used[e] = false;
        int idx[4]; float val[4];
        for (int k = 0; k < 4; k++) {
            int bi = 0; float bv = -INFINITY;
            for (int e = 0; e < NEXP; e++) {
                if (!used[e] && acc[e] > bv) { bv = acc[e]; bi = e; }
            }
            used[bi] = true; idx[k] = bi; val[k] = bv;
        }
        float mx = val[0];
        float g[4]; float s = 0.0f;
        for (int k = 0; k < 4; k++) { g[k] = __expf(val[k] - mx); s += g[k]; }
        float inv = 1.0f / s;
        for (int k = 0; k < 4; k++) {
            float gv = g[k] * inv;
            int e = idx[k];
            atomicAdd(&imp[e], gv);
            atomicAdd(&loadv[e], 1.0f);
            int pos = atomicAdd(&cnt[e], 1);
            etok[e * NTOK + pos] = t;
            egate[e * NTOK + pos] = gv;
        }
    }
}

// ---------------- aux loss ----------------
__global__ void moe_loss_kernel(const float* imp, const float* loadv, float* out) {
    float m1 = 0.0f, m2 = 0.0f;
    for (int e = 0; e < NEXP; e++) { m1 += imp[e]; m2 += loadv[e]; }
    m1 *= (1.0f / NEXP); m2 *= (1.0f / NEXP);
    float v1 = 0.0f, v2 = 0.0f;
    for (int e = 0; e < NEXP; e++) {
        float d1 = imp[e] - m1, d2 = loadv[e] - m2;
        v1 += d1 * d1; v2 += d2 * d2;
    }
    v1 *= (1.0f / (NEXP - 1)); v2 *= (1.0f / (NEXP - 1));
    const float eps = 1e-10f;
    out[640] = 0.01f * (v1 / (m1 * m1 + eps) + v2 / (m2 * m2 + eps));
}

// ---------------- routed expert FFN: f32 WMMA + async LDS gather ----------------
// grid = (expert, token_tile), block = 256 threads (8 waves)
__global__ __launch_bounds__(256)
void moe_ffn_kernel(const float* __restrict__ x,
                    const float* __restrict__ W1, const float* __restrict__ b1,
                    const float* __restrict__ W2, const float* __restrict__ b2,
                    const int* __restrict__ cnt, const int* __restrict__ etok,
                    const float* __restrict__ egate, float* __restrict__ y) {
    __shared__ __align__(16) float sX[16 * XS];
    __shared__ __align__(16) float sH[16 * HS];
    __shared__ int   sTok[16];
    __shared__ float sG[16];

    const int e = blockIdx.x;
    const int tile = blockIdx.y;
    const int count = cnt[e];
    if (tile * 16 >= count) return;   // uniform across the block

    const int tid = threadIdx.x;
    if (tid < 16) {
        int r = tile * 16 + tid;
        if (r < count) { sTok[tid] = etok[e * NTOK + r]; sG[tid] = egate[e * NTOK + r]; }
        else           { sTok[tid] = etok[e * NTOK + tile * 16]; sG[tid] = 0.0f; }
    }
    __syncthreads();

    // gather 16 token rows into LDS via async global->LDS b128 copies (ASYNCcnt path).
    // Each lane: 16B-aligned global address + 16B-aligned LDS byte offset.
    for (int i = tid; i < 16 * (DDIM / 4); i += 256) {
        int r = i >> 8;                 // token row 0..15
        int dc = (i & 255) << 2;        // float column, multiple of 4
        unsigned ldsoff = (unsigned)(uintptr_t)&sX[r * XS + dc];
        unsigned long long gaddr =
            (unsigned long long)(uintptr_t)(x + (size_t)sTok[r] * DDIM + dc);
        asm volatile("global_load_async_to_lds_b128 %0, %1, off"
                     :: "v"(ldsoff), "v"(gaddr) : "memory");
    }
    asm volatile("s_wait_asynccnt 0x0" ::: "memory");
    __syncthreads();

    const int wave   = tid >> 5;
    const int lane   = tid & 31;
    const int lanelo = lane & 15;
    const int khalf  = (lane >> 4) << 1;       // 0 for lanes 0-15, 2 for lanes 16-31
    const int mhi    = (lane >> 4) * 8;        // C-tile row offset

    const float* W1e = W1 + (size_t)e * DDIM * HDIM;
    const float* W2e = W2 + (size_t)e * HDIM * DDIM;

    v8f zero = {};
    v8f acc[8];
#pragma unroll
    for (int t = 0; t < 8; t++) acc[t] = zero;
    const int outbase = wave * 128;

    for (int hc = 0; hc < 8; hc++) {
        const int hbase = hc * 128;
        // ---- phase 1: h tile (16 x 16) for this wave's column slice, K = 1024 ----
        const int hcol = hbase + wave * 16 + lanelo;
        v8f c = zero;
        for (int kt = 0; kt < 256; kt++) {
            const int kb = kt * 4 + khalf;
            v2f a = *(const v2f*)&sX[lanelo * XS + kb];      // A[m][k], m=lanelo
            v2f b;
            b.x = W1e[(size_t)kb * HDIM + hcol];             // B[k][n]
            b.y = W1e[(size_t)(kb + 1) * HDIM + hcol];
            c = __builtin_amdgcn_wmma_f32_16x16x4_f32(false, a, false, b,
                                                      (short)0, c, false, false);
        }
        const float bias1 = b1[e * HDIM + hcol];
#pragma unroll
        for (int j = 0; j < 8; j++) {
            float hv = c[j] + bias1;
            hv = hv > 0.0f ? hv : 0.0f;                      // relu
            sH[(mhi + j) * HS + wave * 16 + lanelo] = hv;
        }
        __syncthreads();

        // ---- phase 2: accumulate this wave's 128 output columns over K = 128 ----
#pragma unroll
        for (int t = 0; t < 8; t++) {
            const int outcol = outbase + t * 16 + lanelo;
            v8f cc = acc[t];
            for (int kt = 0; kt < 32; kt++) {
                const int kb = kt * 4 + khalf;
                v2f a = *(const v2f*)&sH[lanelo * HS + kb];
                v2f b;
                b.x = W2e[(size_t)(hbase + kb) * DDIM + outcol];
                b.y = W2e[(size_t)(hbase + kb + 1) * DDIM + outcol];
                cc = __builtin_amdgcn_wmma_f32_16x16x4_f32(false, a, false, b,
                                                           (short)0, cc, false, false);
            }
            acc[t] = cc;
        }
        __syncthreads();
    }

    // ---- writeout: y[token] += gate * (o + b2) ----
#pragma unroll
    for (int t = 0; t < 8; t++) {
        const int outcol = outbase + t * 16 + lanelo;
        const float bias2 = b2[e * DDIM + outcol];
#pragma unroll
        for (int j = 0; j < 8; j++) {
            const int m = mhi + j;
            float val = sG[m] * (acc[t][j] + bias2);
            atomicAdd(&y[(size_t)sTok[m] * DDIM + outcol], val);
        }
    }
}

// ---------------- sum over M, layernorm, head, losses ----------------
__global__ void moe_post_kernel(const float* __restrict__ y,
                                const float* __restrict__ ln_w, const float* __restrict__ ln_b,
                                const float* __restrict__ W_los, const float* __restrict__ b_los,
                                const float* __restrict__ true_y, float* out) {
    __shared__ float sF[DDIM];
    __shared__ float red[256];
    const int b = blockIdx.x;
    const int tid = threadIdx.x;

    float lsum = 0.0f;
    for (int d = tid; d < DDIM; d += 256) {
        float s = 0.0f;
        const float* yr = y + (size_t)(b * 16) * DDIM + d;
#pragma unroll
        for (int m = 0; m < 16; m++) s += yr[(size_t)m * DDIM];
        sF[d] = s;
        lsum += s;
    }
    red[tid] = lsum; __syncthreads();
    for (int s = 128; s > 0; s >>= 1) { if (tid < s) red[tid] += red[tid + s]; __syncthreads(); }
    const float mu = red[0] * (1.0f / DDIM);
    __syncthreads();

    float vsum = 0.0f;
    for (int d = tid; d < DDIM; d += 256) { float dd = sF[d] - mu; vsum += dd * dd; }
    red[tid] = vsum; __syncthreads();
    for (int s = 128; s > 0; s >>= 1) { if (tid < s) red[tid] += red[tid + s]; __syncthreads(); }
    const float var = red[0] * (1.0f / DDIM);
    const float rstd = rsqrtf(var + 1e-5f);
    __syncthreads();

    for (int d = tid; d < DDIM; d += 256)
        sF[d] = (sF[d] - mu) * rstd * ln_w[d] + ln_b[d];
    __syncthreads();

    float part[10];
#pragma unroll
    for (int o = 0; o < 10; o++) part[o] = 0.0f;
    for (int d = tid; d < DDIM; d += 256) {
        float fv = sF[d];
        const float* wr = W_los + (size_t)d * 10;
#pragma unroll
        for (int o = 0; o < 10; o++) part[o] += fv * wr[o];
    }
    for (int o = 0; o < 10; o++) {
        red[tid] = part[o]; __syncthreads();
        for (int s = 128; s > 0; s >>= 1) { if (tid < s) red[tid] += red[tid + s]; __syncthreads(); }
        if (tid == 0) {
            float sc = red[0] + b_los[o];
            out[b * 10 + o] = sc;
            float diff = sc - true_y[b * 10 + o];
            atomicAdd(&out[641], diff * diff * (1.0f / 640.0f));
        }
        __syncthreads();
    }
}

extern "C" void kernel_launch(void* const* d_in, const int* in_sizes, int n_in,
                              void* d_out, int out_size, void* d_ws, size_t ws_size,
                              hipStream_t stream) {
    const float* x      = (const float*)d_in[0];   // (64,16,1024) -> (1024,1024)
    const float* w_gate = (const float*)d_in[1];   // (1024,16)
    const float* W1     = (const float*)d_in[2];   // (16,1024,1024)
    const float* b1     = (const float*)d_in[3];   // (16,1024)
    const float* W2     = (const float*)d_in[4];   // (16,1024,1024)
    const float* b2     = (const float*)d_in[5];   // (16,1024)
    const float* ln_w   = (const float*)d_in[6];   // (1024,)
    const float* ln_b   = (const float*)d_in[7];   // (1024,)
    const float* W_los  = (const float*)d_in[8];   // (1024,10)
    const float* b_los  = (const float*)d_in[9];   // (10,)
    const float* true_y = (const float*)d_in[10];  // (64,10)
    float* out = (float*)d_out;                    // 640 scores + loss + pred_loss

    // workspace layout
    char* ws = (char*)d_ws;
    float* y     = (float*)ws;                                 // 1024*1024 f32 = 4MB
    float* imp   = (float*)(ws + (size_t)NTOK * DDIM * 4);     // 16
    float* loadv = imp + NEXP;                                 // 16
    int*   cnt   = (int*)(loadv + NEXP);                       // 16
    float* egate = (float*)(cnt + NEXP);                       // 16*1024
    int*   etok  = (int*)(egate + NEXP * NTOK);                // 16*1024

    moe_init_kernel<<<1024, 256, 0, stream>>>(y, imp, loadv, cnt, out);
    moe_gate_kernel<<<NTOK / 8, 256, 0, stream>>>(x, w_gate, imp, loadv, cnt, etok, egate);
    moe_loss_kernel<<<1, 1, 0, stream>>>(imp, loadv, out);
    moe_ffn_kernel<<<dim3(NEXP, NTOK / 16), 256, 0, stream>>>(x, W1, b1, W2, b2,
                                                              cnt, etok, egate, y);
    moe_post_kernel<<<64, 256, 0, stream>>>(y, ln_w, ln_b, W_los, b_los, true_y, out);
}